// Quantizer_53678501266004
// MI455X (gfx1250) — compile-verified
//
#include <hip/hip_runtime.h>
#include <math.h>

typedef float v2f __attribute__((ext_vector_type(2)));
typedef float v8f __attribute__((ext_vector_type(8)));

#define KCODES   512
#define CDIM     64
#define HWDIM    4096          // 64*64
#define LDSROW   66            // padded row stride (floats) to avoid bank conflicts
#define POSB     1024          // positions per block (divides 4096 -> one batch per block)
#define NTHREADS 256
#define QOFF     8388608       // B*C*H*W
#define MIOFF    8388611       // QOFF + 3 scalars

// LDS layout (dynamic):
//   float semb[512*66]   codebook, padded rows
//   float e2s [512]      ||e_k||^2
//   int   zs  [1024]     argmin code index per position in block
//   float wsum[8]        per-wave loss partial sums
__global__ __launch_bounds__(NTHREADS)
void vq_main(const float* __restrict__ x, const float* __restrict__ emb,
             float* __restrict__ out, float* __restrict__ ws) {
  extern __shared__ float lds[];
  float* semb = lds;
  float* e2s  = lds + KCODES * LDSROW;
  int*   zs   = (int*)(e2s + KCODES);
  float* wsum = (float*)(zs + POSB);

  const int tid = threadIdx.x;

  // ---- stage codebook into LDS (coalesced), compute ||e||^2 ----
  for (int i = tid; i < KCODES * CDIM; i += NTHREADS) {
    int r = i >> 6, c = i & 63;
    semb[r * LDSROW + c] = emb[i];
  }
  __syncthreads();
  for (int r = tid; r < KCODES; r += NTHREADS) {
    float s = 0.f;
    #pragma unroll
    for (int c = 0; c < CDIM; ++c) { float v = semb[r * LDSROW + c]; s += v * v; }
    e2s[r] = s;
  }
  __syncthreads();

  const int wave  = tid >> 5;
  const int lane  = tid & 31;
  const int lhalf = lane >> 4;      // 0: K%4 in {0,1}, 1: K%4 in {2,3}
  const int l16   = lane & 15;
  const int gbase = blockIdx.x * POSB;   // global position base; one batch per block

  float wloss = 0.f;

  for (int grp = 0; grp < 8; ++grp) {
    const int m  = gbase + wave * 128 + grp * 16 + l16;  // this lane's row (position)
    const int bb = m >> 12;
    const int pp = m & 4095;
    const float* xp = x + ((size_t)bb << 18) + pp;       // x[b, c, p] with c stride 4096

    // ---- load A fragments: 16 k-steps of 16x4 fp32; accumulate ||x||^2 partials ----
    v2f a[16];
    float px2 = 0.f;
    #pragma unroll
    for (int t = 0; t < 16; ++t) {
      const int c0 = t * 4 + lhalf * 2;
      float ax = xp[(size_t)c0 << 12];
      float ay = xp[(size_t)(c0 + 1) << 12];
      a[t].x = ax; a[t].y = ay;
      px2 += ax * ax + ay * ay;
    }

    float bestv[8];
    int   bestk[8];
    #pragma unroll
    for (int r = 0; r < 8; ++r) { bestv[r] = INFINITY; bestk[r] = 0; }

    // ---- 32 code tiles of 16; D = X . E^T via v_wmma_f32_16x16x4_f32 ----
    for (int T = 0; T < 32; ++T) {
      const int n = T * 16 + l16;                 // this lane's column (code index)
      const float* erow = semb + n * LDSROW;
      v8f acc = {0.f, 0.f, 0.f, 0.f, 0.f, 0.f, 0.f, 0.f};
      #pragma unroll
      for (int t = 0; t < 16; ++t) {
        const int c0 = t * 4 + lhalf * 2;
        v2f bf;
        bf.x = erow[c0];
        bf.y = erow[c0 + 1];
        acc = __builtin_amdgcn_wmma_f32_16x16x4_f32(
            false, a[t], false, bf, (short)0, acc, false, false);
      }
      const float e2v = e2s[n];
      #pragma unroll
      for (int r = 0; r < 8; ++r) {
        float s = e2v - 2.f * acc[r];             // ||e||^2 - 2 x.e  (argmin-equivalent)
        if (s < bestv[r]) { bestv[r] = s; bestk[r] = n; }
      }
    }

    // ---- argmin across the 16 lanes holding a row's columns (tie -> lower index) ----
    #pragma unroll
    for (int mask = 8; mask >= 1; mask >>= 1) {
      #pragma unroll
      for (int r = 0; r < 8; ++r) {
        float ov = __shfl_xor(bestv[r], mask, 32);
        int   ok = __shfl_xor(bestk[r], mask, 32);
        if (ov < bestv[r] || (ov == bestv[r] && ok < bestk[r])) {
          bestv[r] = ov; bestk[r] = ok;
        }
      }
    }
    // lane0 -> rows 0..7, lane16 -> rows 8..15
    if (l16 == 0) {
      const int zb = wave * 128 + grp * 16 + lhalf * 8;
      #pragma unroll
      for (int r = 0; r < 8; ++r) zs[zb + r] = bestk[r];
    }

    // ---- loss partial: sum_rows(||x||^2 + min score) ----
    float lb = 0.f;
    #pragma unroll
    for (int r = 0; r < 8; ++r) lb += bestv[r];
    lb += __shfl_xor(lb, 16, 32);                 // rows 0..15 total (all lanes)
    float ps = px2;
    #pragma unroll
    for (int mask = 16; mask >= 1; mask >>= 1) ps += __shfl_xor(ps, mask, 32);
    if (lane == 0) wloss += ps + lb;
  }

  if (lane == 0) wsum[wave] = wloss;
  __syncthreads();
  if (tid == 0) {
    float s = 0.f;
    #pragma unroll
    for (int w = 0; w < 8; ++w) s += wsum[w];
    ws[blockIdx.x] = s;                            // deterministic per-block sum
  }

  // ---- cooperative coalesced output writes ----
  // min_index_out: flat (B,H,W,C) order -> contiguous run of POSB*64 floats
  float* mi = out + MIOFF + (size_t)gbase * CDIM;
  for (int i = tid; i < POSB * CDIM; i += NTHREADS) {
    mi[i] = semb[zs[i >> 6] * LDSROW + (i & 63)];
  }
  // quantized: (B,C,H,W); per channel a contiguous run of POSB floats
  const int qb = gbase >> 12;
  const int pbase = gbase & 4095;
  float* q = out + ((size_t)qb << 18) + pbase;
  for (int c = 0; c < CDIM; ++c) {
    for (int pos = tid; pos < POSB; pos += NTHREADS) {
      q[((size_t)c << 12) + pos] = semb[zs[pos] * LDSROW + c];
    }
  }
}

__global__ void vq_finalize(const float* __restrict__ ws, float* __restrict__ out) {
  if (threadIdx.x == 0 && blockIdx.x == 0) {
    float s = 0.f;
    for (int i = 0; i < 128; ++i) s += ws[i];
    float mean = s * (1.0f / 8388608.0f);
    out[QOFF + 0] = mean;          // codebook_loss
    out[QOFF + 1] = mean;          // commitment_loss (numerically identical)
    out[QOFF + 2] = 1.2f * mean;   // 0.2*commit + codebook
  }
}

extern "C" void kernel_launch(void* const* d_in, const int* in_sizes, int n_in,
                              void* d_out, int out_size, void* d_ws, size_t ws_size,
                              hipStream_t stream) {
  const float* x   = (const float*)d_in[0];
  const float* emb = (const float*)d_in[1];
  float* out = (float*)d_out;
  float* ws  = (float*)d_ws;

  const size_t shmem = (size_t)(KCODES * LDSROW + KCODES) * sizeof(float)
                     + (size_t)POSB * sizeof(int) + 8 * sizeof(float);
  vq_main<<<128, NTHREADS, shmem, stream>>>(x, emb, out, ws);
  vq_finalize<<<1, 64, 0, stream>>>(ws, out);
}